// GridBasedStructModulePadded_53137335386462
// MI455X (gfx1250) — compile-verified
//
#include <hip/hip_runtime.h>
#include <stdint.h>

// Problem constants (from reference)
#define TOTAL  262144
#define DIM    128
#define NB     64
#define NA     128
#define MAXV   4608

typedef float v2f __attribute__((ext_vector_type(2)));
typedef float v8f __attribute__((ext_vector_type(8)));

// ---------------------------------------------------------------------------
// Kernel 1: logits[t] = dot(E[t,:], W) + b   via V_WMMA_F32_16X16X4_F32
// One wave32 computes a 16-row tile; K=128 accumulated in 32 WMMA steps.
// A (16x4 f32): lane L holds row (L&15); VGPR0/1 = K {0,1} (lanes 0-15) or
// K {2,3} (lanes 16-31).  B (4x16 f32): VGPR0 = row K0 (lanes 0-15) / K2
// (lanes 16-31), VGPR1 = K1/K3. We want only column N=0 nonzero, holding W.
// Key trick: the W elements a lane needs sit at offset 2*hi -- the SAME
// offset pattern as the A fragment -- so the B build is a branchless
// LDS-broadcast load times a per-lane 0/1 selector (no divergence at all).
// D (16x16 f32): VGPR j = M=j (lanes 0-15) / M=8+j (lanes 16-31), N=lane%16,
// so lanes 0 and 16 hold the logit column.
// ---------------------------------------------------------------------------
__global__ __launch_bounds__(256) void logits_wmma_kernel(
    const float* __restrict__ E,
    const float* __restrict__ Ww,
    const float* __restrict__ bw,
    float* __restrict__ logits)
{
    __shared__ float sW[DIM];
    const int tid = threadIdx.x;
    if (tid < DIM) sW[tid] = Ww[tid];
    __syncthreads();

    const float bias = bw[0];
    const int wave = tid >> 5;
    const int lane = tid & 31;
    const int tile = blockIdx.x * 8 + wave;     // 16384 tiles total
    const int rowbase = tile * 16;
    const int m  = lane & 15;
    const int hi = lane >> 4;                   // 0 or 1

    const float*  rowptr = E + (size_t)(rowbase + m) * DIM + 2 * hi;
    const float2* sW2    = reinterpret_cast<const float2*>(sW);
    const float   sel    = (m == 0) ? 1.0f : 0.0f;   // lanes 0 and 16 only

    v8f c = {0.f, 0.f, 0.f, 0.f, 0.f, 0.f, 0.f, 0.f};
    #pragma unroll
    for (int k = 0; k < DIM; k += 4) {
        const float2 av = *reinterpret_cast<const float2*>(rowptr + k);
        const float2 wv = sW2[(k >> 1) + hi];        // LDS broadcast
        v2f a;   a.x  = av.x;        a.y  = av.y;
        v2f bf;  bf.x = wv.x * sel;  bf.y = wv.y * sel;
        c = __builtin_amdgcn_wmma_f32_16x16x4_f32(
                /*neg_a=*/false, a, /*neg_b=*/false, bf,
                /*c_mod=*/(short)0, c, /*reuse_a=*/false, /*reuse_b=*/false);
    }

    if (m == 0) {
        const int mb = rowbase + hi * 8;
        #pragma unroll
        for (int j = 0; j < 8; ++j) logits[mb + j] = c[j] + bias;
    }
}

// ---------------------------------------------------------------------------
// Kernel 2: per-batch segmented stable softmax + weighted coord sum.
// batch_idx is sorted -> binary-search segment bounds. w overwrites logits
// in place (each element touched by exactly one thread, after block syncs).
// ---------------------------------------------------------------------------
__global__ __launch_bounds__(256) void segstats_kernel(
    float* __restrict__ logits,           // in: logits, out: w (in place)
    const float* __restrict__ coords,     // [TOTAL,3]
    const int* __restrict__ bi,           // sorted batch index
    int* __restrict__ segStart,
    int* __restrict__ segCount,
    float* __restrict__ coordB)           // [NB,3]
{
    const int b    = blockIdx.x;
    const int tid  = threadIdx.x;
    const int lane = tid & 31;
    const int wv   = tid >> 5;

    __shared__ int   sStart, sEnd;
    __shared__ float red[8];
    __shared__ float rx[8], ry[8], rz[8];
    __shared__ float sM, sDenom;

    if (tid == 0) {
        int lo = 0, hi = TOTAL;
        while (lo < hi) { int mid = (lo + hi) >> 1; if (bi[mid] < b) lo = mid + 1; else hi = mid; }
        sStart = lo;
        hi = TOTAL;
        while (lo < hi) { int mid = (lo + hi) >> 1; if (bi[mid] < b + 1) lo = mid + 1; else hi = mid; }
        sEnd = lo;
    }
    __syncthreads();
    const int start = sStart, end = sEnd;

    // pass 1: segment max
    float m = -INFINITY;
    for (int t = start + tid; t < end; t += 256) m = fmaxf(m, logits[t]);
    for (int off = 16; off; off >>= 1) m = fmaxf(m, __shfl_xor(m, off, 32));
    if (lane == 0) red[wv] = m;
    __syncthreads();
    if (tid == 0) {
        float v = red[0];
        for (int i = 1; i < 8; ++i) v = fmaxf(v, red[i]);
        sM = (v > -INFINITY) ? v : 0.0f;          // empty-segment guard
    }
    __syncthreads();
    const float mx = sM;

    // pass 2: sum of exp
    float s = 0.0f;
    for (int t = start + tid; t < end; t += 256) s += expf(logits[t] - mx);
    for (int off = 16; off; off >>= 1) s += __shfl_xor(s, off, 32);
    if (lane == 0) red[wv] = s;
    __syncthreads();
    if (tid == 0) {
        float v = 0.0f;
        for (int i = 0; i < 8; ++i) v += red[i];
        sDenom = (v > 0.0f) ? v : 1.0f;
    }
    __syncthreads();
    const float denom = sDenom;

    // pass 3: w = exp(..)/denom (in place), weighted coord accumulation
    float cx = 0.f, cy = 0.f, cz = 0.f;
    for (int t = start + tid; t < end; t += 256) {
        const float wvv = expf(logits[t] - mx) / denom;
        logits[t] = wvv;
        cx += wvv * coords[3 * t + 0];
        cy += wvv * coords[3 * t + 1];
        cz += wvv * coords[3 * t + 2];
    }
    for (int off = 16; off; off >>= 1) {
        cx += __shfl_xor(cx, off, 32);
        cy += __shfl_xor(cy, off, 32);
        cz += __shfl_xor(cz, off, 32);
    }
    if (lane == 0) { rx[wv] = cx; ry[wv] = cy; rz[wv] = cz; }
    __syncthreads();
    if (tid == 0) {
        float x = 0.f, y = 0.f, z = 0.f;
        for (int i = 0; i < 8; ++i) { x += rx[i]; y += ry[i]; z += rz[i]; }
        coordB[3 * b + 0] = x; coordB[3 * b + 1] = y; coordB[3 * b + 2] = z;
        segStart[b] = start; segCount[b] = end - start;
    }
}

// ---------------------------------------------------------------------------
// Kernel 3: broadcast writes (the 151 MB store kernel).
// grid = (NB, 16); each block stages the padded weight row in LDS (18 KB)
// using GLOBAL_LOAD_ASYNC_TO_LDS_B32 (ASYNCcnt path), zero-fills the tail
// with DS stores, then writes 8 atom rows with float4 (b128) stores.
// Unmasked rows get zeros. Handles the reference's pos=min(pos,MAXV-1)
// overflow clobber.
// ---------------------------------------------------------------------------
__global__ __launch_bounds__(256) void broadcast_kernel(
    const float* __restrict__ w,
    const int* __restrict__ mask,         // [NB,NA]
    const int* __restrict__ segStart,
    const int* __restrict__ segCount,
    const float* __restrict__ coordB,
    float* __restrict__ outPred,          // [NB,NA,3]
    float* __restrict__ outAttn)          // [NB,NA,MAXV]
{
    __shared__ float4 row4[MAXV / 4];
    float* row = reinterpret_cast<float*>(row4);

    const int b   = blockIdx.x;
    const int tid = threadIdx.x;
    const int start = segStart[b];
    const int count = segCount[b];
    const int cmax  = (count < MAXV) ? count : MAXV;

    // Async global -> LDS copy of the live weights (per-lane b32 transfers).
    // Flat LDS addresses use addr[31:0] as the wave-relative LDS offset, so
    // the low 32 bits of the generic shared pointer are the VDST operand.
    for (int v = tid; v < cmax; v += 256) {
        const uint32_t lds_off = (uint32_t)(uintptr_t)(row + v);
        const uint64_t gaddr   = (uint64_t)(uintptr_t)(w + start + v);
        asm volatile("global_load_async_to_lds_b32 %0, %1, off"
                     :: "v"(lds_off), "v"(gaddr) : "memory");
    }
    // Zero tail [cmax, MAXV) with normal DS stores (disjoint from async region).
    for (int v = cmax + tid; v < MAXV; v += 256) row[v] = 0.0f;

    // predicted_coords (tiny) while the async copy is in flight
    if (blockIdx.y == 0 && tid < NA) {
        const float mk = (mask[b * NA + tid] != 0) ? 1.0f : 0.0f;
        const size_t p = (size_t)(b * NA + tid) * 3;
        outPred[p + 0] = coordB[3 * b + 0] * mk;
        outPred[p + 1] = coordB[3 * b + 1] * mk;
        outPred[p + 2] = coordB[3 * b + 2] * mk;
    }

    asm volatile("s_wait_asynccnt 0" ::: "memory");
    // Overflow clobber (after async wait: overrides row[MAXV-1])
    if (count > MAXV && tid == 0) row[MAXV - 1] = w[start + count - 1];
    __syncthreads();

    const float4 zero4 = make_float4(0.f, 0.f, 0.f, 0.f);
    #pragma unroll
    for (int ai = 0; ai < 8; ++ai) {
        const int a = blockIdx.y * 8 + ai;
        const bool mk = (mask[b * NA + a] != 0);
        float4* dst = reinterpret_cast<float4*>(outAttn + (size_t)(b * NA + a) * MAXV);
        for (int v = tid; v < MAXV / 4; v += 256) {
            dst[v] = mk ? row4[v] : zero4;
        }
    }
}

// ---------------------------------------------------------------------------
extern "C" void kernel_launch(void* const* d_in, const int* in_sizes, int n_in,
                              void* d_out, int out_size, void* d_ws, size_t ws_size,
                              hipStream_t stream) {
    const float* E      = (const float*)d_in[0];   // virtual_embeddings [TOTAL,DIM]
    const float* coords = (const float*)d_in[1];   // virtual_coords    [TOTAL,3]
    const int*   mask   = (const int*)d_in[2];     // target_mask       [NB,NA]
    const int*   bi     = (const int*)d_in[3];     // batch_idx (sorted) [TOTAL]
    const float* Ww     = (const float*)d_in[4];   // W_w [1,DIM]
    const float* bw     = (const float*)d_in[5];   // b_w [1]

    float* out     = (float*)d_out;
    float* outPred = out;                          // NB*NA*3
    float* outAttn = out + (size_t)NB * NA * 3;    // NB*NA*MAXV

    float* ws      = (float*)d_ws;
    float* logits  = ws;                           // TOTAL floats, becomes w
    float* coordB  = ws + TOTAL;                   // NB*3
    int*   segS    = (int*)(ws + TOTAL + NB * 3);  // NB
    int*   segC    = segS + NB;                    // NB

    logits_wmma_kernel<<<dim3(TOTAL / (16 * 8)), 256, 0, stream>>>(E, Ww, bw, logits);
    segstats_kernel<<<dim3(NB), 256, 0, stream>>>(logits, coords, bi, segS, segC, coordB);
    broadcast_kernel<<<dim3(NB, 16), 256, 0, stream>>>(logits, mask, segS, segC, coordB,
                                                       outPred, outAttn);
}